// PatchManifoldLossPre_41472204210222
// MI455X (gfx1250) — compile-verified
//
#include <hip/hip_runtime.h>
#include <hip/hip_bf16.h>

// PatchManifoldLoss on MI455X (gfx1250):
//   value = mean_{a,b}( weights[a,b] * mean_n (P[a,n]-P[b,n])^2 ) with
//   P = 16 x N (N = 152*144) from antialiased bilinear 512->48 resize.
// Strategy: separable antialiased resize (memory-bound, 159MB read ~= 6.8us
// at 23.3 TB/s), then Gram matrix G = P P^T via V_WMMA_F32_16X16X4_F32 in
// exact fp32, then msd[a,b] = (G[aa]+G[bb]-2G[ab])/N.
// Workspace use: ~15 MB (tmp of H-pass) + small tables.

typedef __attribute__((ext_vector_type(2))) float v2f;
typedef __attribute__((ext_vector_type(8))) float v8f;

#define IN_HW 512
#define OUT_HW 48
#define PATCH 12
#define NPOS 144          // 12*12 positions per image per patch
#define MAXTAPS 24        // triangle support is <= 22 taps; pad to 24

// workspace layout (bytes)
#define WS_G      0       // 256 floats  (16x16 Gram accumulator)
#define WS_STARTS 1024    // 48 ints
#define WS_NTAPS  1216    // 48 ints
#define WS_WTS    1408    // 48*24 floats
#define WS_TMP    8192    // nimg*48*512 floats (~14.9 MB)

// ---------------------------------------------------------------- init ----
// Zero the Gram accumulator and build the 1D antialiased-triangle tap table
// (replicates jax.image.resize compute_weight_mat for scale 48/512).
__global__ void pml_init(float* __restrict__ G, int* __restrict__ starts,
                         int* __restrict__ ntaps, float* __restrict__ wts) {
  int tid = threadIdx.x;
  if (tid < 256) G[tid] = 0.0f;
  if (tid < OUT_HW) {
    const float inv = (float)IN_HW / (float)OUT_HW;   // 10.6667 = kernel_scale
    float sample = (tid + 0.5f) * inv - 0.5f;         // half-pixel centers
    int j0 = (int)ceilf(sample - inv);
    if (j0 < 0) j0 = 0;
    int j1 = (int)floorf(sample + inv);
    if (j1 > IN_HW - 1) j1 = IN_HW - 1;
    int n = j1 - j0 + 1;
    if (n > MAXTAPS) n = MAXTAPS;
    float w[MAXTAPS];
    float total = 0.0f;
    for (int t = 0; t < MAXTAPS; ++t) {
      float x = fabsf(sample - (float)(j0 + t)) * (1.0f / inv);
      float v = (t < n) ? fmaxf(0.0f, 1.0f - x) : 0.0f;
      w[t] = v;
      total += v;
    }
    float rtot = 1.0f / total;                        // edge renormalization
    for (int t = 0; t < MAXTAPS; ++t) wts[tid * MAXTAPS + t] = w[t] * rtot;
    starts[tid] = j0;
    ntaps[tid] = n;
  }
}

// -------------------------------------------------------------- H-pass ----
// tmp[img][r][c] = sum_t wts[r][t] * pred[img][starts[r]+t][c]
// One block per (img, output-row); coalesced along c. Reads the full 159MB
// input (the roofline term), writes 15MB.
__global__ void pml_hpass(const float* __restrict__ pred,
                          const int* __restrict__ starts,
                          const int* __restrict__ ntaps,
                          const float* __restrict__ wts,
                          float* __restrict__ tmp) {
  int img = blockIdx.x / OUT_HW;
  int r = blockIdx.x - img * OUT_HW;
  const float* src = pred + (size_t)img * (IN_HW * IN_HW);
  float* dst = tmp + ((size_t)img * OUT_HW + r) * IN_HW;
  int j0 = starts[r];          // scalar (SGPR) — uniform per block
  int n = ntaps[r];
  const float* w = wts + r * MAXTAPS;
  for (int c = threadIdx.x; c < IN_HW; c += blockDim.x) {
    const float* p = src + (size_t)j0 * IN_HW + c;
    __builtin_prefetch(p + (size_t)n * IN_HW, 0, 0);  // global_prefetch_b8
    float acc = 0.0f;
    for (int t = 0; t < n; ++t)
      acc = fmaf(w[t], p[(size_t)t * IN_HW], acc);
    dst[c] = acc;
  }
}

// ------------------------------------------------- W-pass + WMMA Gram ----
// One block (8 wave32) per image. Phase 1: resized 48x48 tile into LDS.
// Phase 2: G += P_img * P_img^T with v_wmma_f32_16x16x4_f32; for a Gram,
// the A (16x4, lane=M, vgpr=K) and B (4x16, lane=N, vgpr=K) VGPR layouts
// carry identical values, so one v2f per lane serves as both operands.
__global__ void pml_wpass_gram(const float* __restrict__ tmp,
                               const int* __restrict__ starts,
                               const int* __restrict__ ntaps,
                               const float* __restrict__ wts,
                               float* __restrict__ G) {
  __shared__ float rs[OUT_HW * OUT_HW];   // resized tile, 9 KB
  __shared__ float Gacc[256];             // per-block 16x16 accumulator
  int tid = threadIdx.x;
  int img = blockIdx.x;
  Gacc[tid] = 0.0f;

  const float* base = tmp + (size_t)img * OUT_HW * IN_HW;
  for (int e = tid; e < OUT_HW * OUT_HW; e += 256) {
    int r = e / OUT_HW;
    int cc = e - r * OUT_HW;
    int j0 = starts[cc];
    int n = ntaps[cc];
    const float* w = wts + cc * MAXTAPS;
    const float* p = base + (size_t)r * IN_HW + j0;
    float acc = 0.0f;
    for (int t = 0; t < n; ++t) acc = fmaf(w[t], p[t], acc);
    rs[e] = acc;
  }
  __syncthreads();

  // Wave-level WMMA over K-chunks of 4 flattened positions (36 chunks).
  int wv = tid >> 5;
  int lane = tid & 31;
  int l = lane & 15;       // = M for A, = N for B (identical for Gram)
  int hi = lane >> 4;      // lanes 16-31 hold K=2,3
  int prow = (l >> 2) * PATCH;   // patch row origin for p = l
  int pcol = (l & 3) * PATCH;    // patch col origin
  v8f c = {0.f, 0.f, 0.f, 0.f, 0.f, 0.f, 0.f, 0.f};
  for (int t = wv; t < NPOS / 4; t += 8) {   // wave-uniform trip count
    int nA = 4 * t + 2 * hi;                 // position for VGPR0 (K = 2*hi)
    int hrA = nA / PATCH, wcA = nA - hrA * PATCH;
    int nB = nA + 1;                         // position for VGPR1 (K = 2*hi+1)
    int hrB = nB / PATCH, wcB = nB - hrB * PATCH;
    v2f a;
    a.x = rs[(prow + hrA) * OUT_HW + pcol + wcA];
    a.y = rs[(prow + hrB) * OUT_HW + pcol + wcB];
    // D = A(16x4) * B(4x16) + C, all fp32 (exact Gram accumulation)
    c = __builtin_amdgcn_wmma_f32_16x16x4_f32(
        /*neg_a=*/false, a, /*neg_b=*/false, a,
        /*c_mod=*/(short)0, c, /*reuse_a=*/false, /*reuse_b=*/false);
  }
  // D layout: VGPR v, lanes 0-15 -> M=v, lanes 16-31 -> M=v+8; N = l.
  for (int v = 0; v < 8; ++v)
    atomicAdd(&Gacc[(v + 8 * hi) * 16 + l], c[v]);   // ds_add_f32
  __syncthreads();
  atomicAdd(&G[tid], Gacc[tid]);                     // 256 global atomics
}

// ------------------------------------------------------------ finalize ----
__global__ void pml_final(const float* __restrict__ G,
                          const float* __restrict__ wgt,
                          float* __restrict__ out, float invN) {
  __shared__ float red[32];
  int lane = threadIdx.x;
  float part = 0.0f;
  for (int i = 0; i < 8; ++i) {
    int e = lane * 8 + i;          // 0..255
    int a = e >> 4, b = e & 15;
    float msd = (G[a * 17] + G[b * 17] - 2.0f * G[a * 16 + b]) * invN;
    part += wgt[e] * msd;
  }
  red[lane] = part;
  __syncthreads();
  if (lane == 0) {
    float s = 0.0f;
    for (int i = 0; i < 32; ++i) s += red[i];
    out[0] = s * (1.0f / 256.0f);  // mean over 16x16, ALPHA = 1
  }
}

// ---------------------------------------------------------------- host ----
extern "C" void kernel_launch(void* const* d_in, const int* in_sizes, int n_in,
                              void* d_out, int out_size, void* d_ws, size_t ws_size,
                              hipStream_t stream) {
  const float* pred = (const float*)d_in[0];   // [8,19,512,512] fp32
  const float* wgt  = (const float*)d_in[1];   // [16,16] fp32
  float* out = (float*)d_out;                  // scalar fp32

  char* ws = (char*)d_ws;                      // needs ~15 MB
  float* G      = (float*)(ws + WS_G);
  int*   starts = (int*)  (ws + WS_STARTS);
  int*   ntaps  = (int*)  (ws + WS_NTAPS);
  float* wts    = (float*)(ws + WS_WTS);
  float* tmp    = (float*)(ws + WS_TMP);

  int nimg = in_sizes[0] / (IN_HW * IN_HW);    // 8*19 = 152

  pml_init<<<1, 256, 0, stream>>>(G, starts, ntaps, wts);
  pml_hpass<<<nimg * OUT_HW, 256, 0, stream>>>(pred, starts, ntaps, wts, tmp);
  pml_wpass_gram<<<nimg, 256, 0, stream>>>(tmp, starts, ntaps, wts, G);
  float invN = 1.0f / (float)(nimg * NPOS);
  pml_final<<<1, 32, 0, stream>>>(G, wgt, out, invN);
}